// BilinearResampling_5050881540165
// MI455X (gfx1250) — compile-verified
//
#include <hip/hip_runtime.h>
#include <hip/hip_bf16.h>

#define C_DIM 128
#define H_DIM 376
#define W_DIM 1248
#define HW_DIM (H_DIM * W_DIM)
#define BLOCK_T 256
#define C_SPLIT 4
#define C_PER_BLOCK (C_DIM / C_SPLIT)
#define PIPE_STAGES 4

// ---- CDNA5 async global->LDS path detection (device pass only) ----
#if defined(__gfx1250__) && __has_builtin(__builtin_amdgcn_global_load_async_to_lds_b64)
#define USE_ASYNC 1
#else
#define USE_ASYNC 0
#endif

#if defined(__gfx1250__)
#if __has_builtin(__builtin_amdgcn_s_wait_asynccnt)
#define WAIT_ASYNC(n) do { __builtin_amdgcn_s_wait_asynccnt(n); asm volatile("" ::: "memory"); } while (0)
#else
#define WAIT_ASYNC(n) asm volatile("s_wait_asynccnt %0" :: "i"(n) : "memory")
#endif
#else
#define WAIT_ASYNC(n) do {} while (0)
#endif

#if USE_ASYNC
// Exact pointee type from the builtin's signature (v2i in AS1 / AS3).
typedef int v2i __attribute__((vector_size(2 * sizeof(int))));
typedef __attribute__((address_space(1))) v2i GV2;   // global v2i
typedef __attribute__((address_space(3))) v2i LV2;   // LDS v2i
#define ASYNC_LD_B64(gp, lp) \
    __builtin_amdgcn_global_load_async_to_lds_b64((GV2*)(gp), (LV2*)(lp), 0, 0)
#endif

__global__ __launch_bounds__(BLOCK_T) void BilinearResampling_kernel(
    const float* __restrict__ fm,   // [C,H,W]
    const float* __restrict__ uv,   // [N,2]
    const int*   __restrict__ dsc,  // scalar downscale
    float*       __restrict__ out,  // [C,N]
    int N)
{
#if USE_ASYNC
    __shared__ __align__(16) float sbuf[PIPE_STAGES][BLOCK_T][4];
#endif
    const int t = threadIdx.x;
    const int n = blockIdx.x * BLOCK_T + t;
    if (n >= N) return;

    const float inv_ds = 1.0f / (float)dsc[0];
    const float u = uv[2 * n + 0] * inv_ds;
    const float v = uv[2 * n + 1] * inv_ds;
    int ulo = (int)u;
    int vlo = (int)v;
    ulo = min(max(ulo, 0), W_DIM - 2);
    vlo = min(max(vlo, 0), H_DIM - 2);
    const float du = u - (float)ulo;
    const float dv = v - (float)vlo;
    const float w00 = (1.0f - dv) * (1.0f - du);   // f[vlo][ulo]
    const float w01 = (1.0f - dv) * du;            // f[vlo][uhi]
    const float w10 = dv * (1.0f - du);            // f[vhi][ulo]
    const float w11 = dv * du;                     // f[vhi][uhi]

    const int c0 = blockIdx.y * C_PER_BLOCK;
    const float* base = fm + (size_t)c0 * HW_DIM + (size_t)vlo * W_DIM + (size_t)ulo;
    float* op = out + (size_t)c0 * N + (size_t)n;

#if USE_ASYNC
    // --- 4-deep software pipeline via async global->LDS (ASYNCcnt-tracked) ---
    // Stage for channel i lives in sbuf[i&3][t][0..3] = {f00,f01,f10,f11}.
    // Steady state: stages i..i+3 outstanding (8 b64 loads); async loads
    // complete in order, so wait asynccnt<=6 retires exactly stage i.
#define ISSUE_STAGE(j) do {                                       \
        const float* _nb = base + (size_t)(j) * HW_DIM;           \
        ASYNC_LD_B64(_nb,         &sbuf[(j) & 3][t][0]);          \
        ASYNC_LD_B64(_nb + W_DIM, &sbuf[(j) & 3][t][2]);          \
    } while (0)

#define CONSUME_STAGE(i) do {                                                          \
        const float4 _f = *(const float4*)&sbuf[(i) & 3][t][0];                        \
        const float _r = fmaf(_f.x, w00, fmaf(_f.y, w01, fmaf(_f.z, w10, _f.w * w11)));\
        __builtin_nontemporal_store(_r, op + (size_t)(i) * N);                         \
    } while (0)

    ISSUE_STAGE(0);
    ISSUE_STAGE(1);
    ISSUE_STAGE(2);

    #pragma unroll 4
    for (int i = 0; i < C_PER_BLOCK - 3; ++i) {
        ISSUE_STAGE(i + 3);
        WAIT_ASYNC(6);      // oldest stage (i) complete
        CONSUME_STAGE(i);
    }
    WAIT_ASYNC(4); CONSUME_STAGE(C_PER_BLOCK - 3);
    WAIT_ASYNC(2); CONSUME_STAGE(C_PER_BLOCK - 2);
    WAIT_ASYNC(0); CONSUME_STAGE(C_PER_BLOCK - 1);

#undef ISSUE_STAGE
#undef CONSUME_STAGE
#else
    // --- direct VGPR gather fallback, with CDNA5 global_prefetch_b8 lookahead ---
    #pragma unroll 4
    for (int i = 0; i < C_PER_BLOCK; ++i) {
        const float* p = base + (size_t)i * HW_DIM;
        if (i + 4 < C_PER_BLOCK) {
            __builtin_prefetch(p + 4 * (size_t)HW_DIM, 0, 0);
            __builtin_prefetch(p + 4 * (size_t)HW_DIM + W_DIM, 0, 0);
        }
        const float f00 = p[0];
        const float f01 = p[1];
        const float f10 = p[W_DIM];
        const float f11 = p[W_DIM + 1];
        const float r = fmaf(f00, w00, fmaf(f01, w01, fmaf(f10, w10, f11 * w11)));
        __builtin_nontemporal_store(r, op + (size_t)i * N);
    }
#endif
}

extern "C" void kernel_launch(void* const* d_in, const int* in_sizes, int n_in,
                              void* d_out, int out_size, void* d_ws, size_t ws_size,
                              hipStream_t stream) {
    const float* fm  = (const float*)d_in[0];   // [C,H,W] f32
    const float* uv  = (const float*)d_in[1];   // [N,2]   f32
    const int*   dsc = (const int*)d_in[2];     // scalar int
    float*       out = (float*)d_out;           // [C,N]   f32
    const int N = in_sizes[1] / 2;

    dim3 grid((N + BLOCK_T - 1) / BLOCK_T, C_SPLIT);
    dim3 block(BLOCK_T);
    hipLaunchKernelGGL(BilinearResampling_kernel, grid, block, 0, stream,
                       fm, uv, dsc, out, N);
}